// QuantizedHyperbolicDistanceModule_74053826118129
// MI455X (gfx1250) — compile-verified
//
#include <hip/hip_runtime.h>
#include <hip/hip_bf16.h>

typedef __attribute__((ext_vector_type(2))) float v2f;
typedef __attribute__((ext_vector_type(4))) float v4f;
typedef __attribute__((ext_vector_type(8))) float v8f;

// Problem constants (from reference): B=2,H=8,N=2048,D=64
constexpr int N  = 2048;
constexpr int D  = 64;
constexpr int BH = 16;

#define CURV      1.0f
#define EPS_DENOM 1e-6f

// ---------------------------------------------------------------------------
// Pass 1: row squared-norms for q and k -> d_ws (qn at ws[0], kn at ws[BH*N])
// 65536 threads, one row (64 floats) each.
// ---------------------------------------------------------------------------
__global__ __launch_bounds__(256) void row_norm_kernel(
    const float* __restrict__ q, const float* __restrict__ k,
    float* __restrict__ ws) {
  const int tid = blockIdx.x * 256 + threadIdx.x;  // [0, 2*BH*N)
  const bool is_k = tid >= BH * N;
  const int row = is_k ? (tid - BH * N) : tid;
  const float* src = (is_k ? k : q) + (size_t)row * D;
  float s = 0.0f;
#pragma unroll
  for (int d = 0; d < D; d += 4) {
    v4f v = *(const v4f*)(src + d);
    s += v.x * v.x + v.y * v.y + v.z * v.z + v.w * v.w;
  }
  ws[tid] = s;  // qn rows then kn rows, contiguous
}

// ---------------------------------------------------------------------------
// Pass 2: each wave computes a 16(i) x 64(j) strip of one (b,h) output.
// Per (b,h): 128 i-tiles * 32 j-strips = 4096 strips; total 65536 strips.
// ---------------------------------------------------------------------------
template <bool USE_WS>
__global__ __launch_bounds__(256) void hyperbolic_dist_kernel(
    const float* __restrict__ q, const float* __restrict__ k,
    const float* __restrict__ qn_ws, const float* __restrict__ kn_ws,
    float* __restrict__ out) {
  const int lane   = threadIdx.x & 31;
  const int waveId = threadIdx.x >> 5;
  const int strip  = blockIdx.x * 8 + waveId;

  const int bh  = strip >> 12;          // strip / 4096
  const int rem = strip & 4095;
  const int i0  = (rem >> 5) << 4;      // (rem/32)*16
  const int j0  = (rem & 31) << 6;      // (rem%32)*64

  const float* qbase = q + ((size_t)bh * N + i0) * D;  // 16 rows of q
  const float* kbase = k + ((size_t)bh * N + j0) * D;  // 64 rows of k

  const int m     = lane & 15;   // M (for A) / N (for B) index
  const int khalf = lane >> 4;   // which K-pair this half-wave holds

  // ---- GEMM: qk[i,j] over K=64 via 16 x V_WMMA_F32_16X16X4_F32 per tile ----
  v8f acc[4];
  acc[0] = v8f{}; acc[1] = v8f{}; acc[2] = v8f{}; acc[3] = v8f{};

#pragma unroll
  for (int kk = 0; kk < D; kk += 4) {
    // A (16x4 f32): lane holds q[m][kk + 2*khalf + {0,1}]
    v2f a = *(const v2f*)(qbase + m * D + kk + 2 * khalf);
#pragma unroll
    for (int jj = 0; jj < 4; ++jj) {
      // B (4x16 f32): lane holds k[j0+16*jj+m][kk + 2*khalf + {0,1}]
      v2f b = *(const v2f*)(kbase + (16 * jj + m) * D + kk + 2 * khalf);
      acc[jj] = __builtin_amdgcn_wmma_f32_16x16x4_f32(
          /*neg_a=*/false, a, /*neg_b=*/false, b,
          /*c_mod=*/(short)0, acc[jj], /*reuse_a=*/false, /*reuse_b=*/false);
    }
  }

  // ---- Norm acquisition ----
  // qn: lane p holds |q_row(i0 + (p&15))|^2
  // kn_lo: lane p holds |k_row(j0 + p)|^2 ; kn_hi: |k_row(j0 + 32 + p)|^2
  float qn, kn_lo, kn_hi;
  if constexpr (USE_WS) {
    qn    = qn_ws[(size_t)bh * N + i0 + (lane & 15)];
    kn_lo = kn_ws[(size_t)bh * N + j0 + lane];
    kn_hi = kn_ws[(size_t)bh * N + j0 + 32 + lane];
  } else {
    qn = 0.0f;
    {
      const float* r = qbase + (lane & 15) * D;
#pragma unroll
      for (int d = 0; d < D; d += 4) {
        v4f v = *(const v4f*)(r + d);
        qn += v.x * v.x + v.y * v.y + v.z * v.z + v.w * v.w;
      }
    }
    float kna = 0.0f, knb = 0.0f;  // cols 2*lane, 2*lane+1
    {
      const float* ra = kbase + (2 * lane) * D;
      const float* rb = kbase + (2 * lane + 1) * D;
#pragma unroll
      for (int d = 0; d < D; d += 4) {
        v4f va = *(const v4f*)(ra + d);
        v4f vb = *(const v4f*)(rb + d);
        kna += va.x * va.x + va.y * va.y + va.z * va.z + va.w * va.w;
        knb += vb.x * vb.x + vb.y * vb.y + vb.z * vb.z + vb.w * vb.w;
      }
    }
    // repack interleaved (2p, 2p+1) -> linear (lane), (32+lane)
    float lo_a = __shfl(kna, lane >> 1, 32);
    float lo_b = __shfl(knb, lane >> 1, 32);
    float hi_a = __shfl(kna, 16 + (lane >> 1), 32);
    float hi_b = __shfl(knb, 16 + (lane >> 1), 32);
    kn_lo = (lane & 1) ? lo_b : lo_a;
    kn_hi = (lane & 1) ? hi_b : hi_a;
  }

  // ---- Epilogue: acosh(1 + 2*c*d2/denom) ----
  // C/D layout: lane -> N = lane&15 ; VGPR r -> M = r + 8*(lane>>4)
#pragma unroll
  for (int jj = 0; jj < 4; ++jj) {
    const int c = jj * 16 + m;                       // column within strip
    const float knsrc = (jj < 2) ? kn_lo : kn_hi;    // jj compile-time
    const float knc = __shfl(knsrc, c & 31, 32);
    const float omk = 1.0f - CURV * knc;

    float* orow = out + ((size_t)bh * N + i0) * N + (size_t)(j0 + c);
#pragma unroll
    for (int r = 0; r < 8; ++r) {
      const int M = r + 8 * khalf;
      float qnm = __shfl(qn, M, 32);                 // |q_row(i0+M)|^2
      float qkv = acc[jj][r];

      float dsq   = fmaxf(qnm + knc - 2.0f * qkv, 0.0f);
      float denom = fmaxf((1.0f - CURV * qnm) * omk, EPS_DENOM);
      float carg  = 1.0f + 2.0f * CURV * dsq * __builtin_amdgcn_rcpf(denom);
      carg = fmaxf(carg, 1.0f);
      // acosh(x) = log(x + sqrt(x^2 - 1))
      float res = __logf(carg + sqrtf(carg * carg - 1.0f));

      orow[(size_t)M * N] = res;
    }
  }
}

extern "C" void kernel_launch(void* const* d_in, const int* in_sizes, int n_in,
                              void* d_out, int out_size, void* d_ws, size_t ws_size,
                              hipStream_t stream) {
  (void)in_sizes; (void)n_in; (void)out_size;
  const float* q = (const float*)d_in[0];
  const float* k = (const float*)d_in[1];
  float* out     = (float*)d_out;

  const size_t ws_needed = (size_t)2 * BH * N * sizeof(float);  // 256 KB
  dim3 grid(BH * (N / 16) * (N / 64) / 8);  // 8192 blocks, 8 waves each
  dim3 block(256);

  if (d_ws != nullptr && ws_size >= ws_needed) {
    float* qn_ws = (float*)d_ws;
    float* kn_ws = qn_ws + (size_t)BH * N;
    hipLaunchKernelGGL(row_norm_kernel, dim3(2 * BH * N / 256), block, 0,
                       stream, q, k, (float*)d_ws);
    hipLaunchKernelGGL(hyperbolic_dist_kernel<true>, grid, block, 0, stream,
                       q, k, qn_ws, kn_ws, out);
  } else {
    hipLaunchKernelGGL(hyperbolic_dist_kernel<false>, grid, block, 0, stream,
                       q, k, (const float*)nullptr, (const float*)nullptr, out);
  }
}